// GATGuard_32950989094962
// MI455X (gfx1250) — compile-verified
//
#include <hip/hip_runtime.h>

typedef float v2f __attribute__((ext_vector_type(2)));
typedef float v8f __attribute__((ext_vector_type(8)));

#define LRELU_ATT 0.2f
#define LRELU_ACT 0.01f
#define SIM_TH 0.1f

__device__ __forceinline__ float leakyf(float v, float s) { return v > 0.f ? v : v * s; }
// order-preserving float<->int bijection for signed atomicMax
__device__ __forceinline__ int f2o(float f) { int i = __float_as_int(f); return i >= 0 ? i : (i ^ 0x7fffffff); }
__device__ __forceinline__ float o2f(int i) { return __int_as_float(i >= 0 ? i : (i ^ 0x7fffffff)); }
__device__ __forceinline__ float wred32(float v) {
#pragma unroll
  for (int o = 16; o > 0; o >>= 1) v += __shfl_xor(v, o, 32);
  return v;
}

// ---------------- fp32 WMMA GEMM: C[M,Nc] = A[M,K] @ B[K,Nc] (row-major) ----
// One wave -> one 16x16 tile; block = 4 waves covering 4 consecutive n-tiles.
__global__ void gg_gemm_f32_wmma(const float* __restrict__ A, const float* __restrict__ B,
                                 float* __restrict__ C, int M, int K, int Nc) {
  const int wave = threadIdx.x >> 5;
  const int lane = threadIdx.x & 31;
  const int m0 = blockIdx.x * 16;
  const int n0 = (blockIdx.y * 4 + wave) * 16;
  if (m0 >= M || n0 >= Nc) return;          // wave-uniform: EXEC stays all-ones
  const int half = lane >> 4;               // lane group 0..15 / 16..31
  const int r = lane & 15;
  const int arow = min(m0 + r, M - 1);
  v8f acc = {};
  for (int k = 0; k < K; k += 4) {
    const int ka = k + 2 * half;            // A 16x4: lanes 0-15 K={k,k+1}, 16-31 K={k+2,k+3}
    v2f a, b;
    a.x = A[(size_t)arow * K + ka];
    a.y = A[(size_t)arow * K + ka + 1];
    b.x = B[(size_t)ka * Nc + n0 + r];      // B 4x16 mirrored layout
    b.y = B[(size_t)(ka + 1) * Nc + n0 + r];
    acc = __builtin_amdgcn_wmma_f32_16x16x4_f32(false, a, false, b, (short)0, acc,
                                                false, false);
  }
#pragma unroll
  for (int v = 0; v < 8; ++v) {             // C: VGPR v -> M = v (+8 for upper lanes)
    const int row = m0 + v + 8 * half;
    if (row < M) C[(size_t)row * Nc + n0 + r] = acc[v];
  }
}

// ---------------- row L2 norms (wave per row) -------------------------------
__global__ void gg_row_norm(const float* __restrict__ X, float* __restrict__ nrm,
                            int N, int Kd) {
  const int row = blockIdx.x * (blockDim.x >> 5) + (threadIdx.x >> 5);
  const int lane = threadIdx.x & 31;
  if (row >= N) return;
  float s = 0.f;
  for (int j = lane; j < Kd; j += 32) { float v = X[(size_t)row * Kd + j]; s += v * v; }
  s = wred32(s);
  if (lane == 0) nrm[row] = fmaxf(sqrtf(s), 1e-12f);
}

// ---------------- per-edge cosine-sim mask (wave per edge) ------------------
__global__ void gg_edge_mask(const float* __restrict__ X, const float* __restrict__ nrm,
                             const int* __restrict__ src, const int* __restrict__ dst,
                             int* __restrict__ mask, int E, int Kd, int combine) {
  const int e = blockIdx.x * (blockDim.x >> 5) + (threadIdx.x >> 5);
  const int lane = threadIdx.x & 31;
  if (e >= E) return;
  if (combine && mask[e] == 0) return;      // cumulative pruning
  const int s = src[e], d = dst[e];
  float acc = 0.f;
  for (int j = lane; j < Kd; j += 32)
    acc += X[(size_t)s * Kd + j] * X[(size_t)d * Kd + j];
  acc = wred32(acc);
  if (lane == 0) {
    const float cs = acc / (nrm[s] * nrm[d]);
    mask[e] = (cs >= SIM_TH) ? 1 : 0;
  }
}

// ---------------- attention logits + self-loop max init ---------------------
__global__ void gg_logits(const float* __restrict__ Hf, const float* __restrict__ al,
                          const float* __restrict__ ar, float* __restrict__ el,
                          float* __restrict__ er, int* __restrict__ mint,
                          int N, int H, int D) {
  const int idx = blockIdx.x * blockDim.x + threadIdx.x;
  if (idx >= N * H) return;
  const int n = idx / H, h = idx % H;
  const float* row = Hf + (size_t)n * H * D + (size_t)h * D;
  const float* a = al + h * D;
  const float* b = ar + h * D;
  float e1 = 0.f, e2 = 0.f;
  for (int d = 0; d < D; ++d) { const float v = row[d]; e1 += v * a[d]; e2 += v * b[d]; }
  el[idx] = e1; er[idx] = e2;
  mint[idx] = f2o(leakyf(e1 + e2, LRELU_ATT));   // self loop is always present
}

// ---------------- segment max over dst (thread per edge-head) ---------------
__global__ void gg_max_edges(const float* __restrict__ el, const float* __restrict__ er,
                             const int* __restrict__ src, const int* __restrict__ dst,
                             const int* __restrict__ mask, int* __restrict__ mint,
                             int E, int H) {
  const int idx = blockIdx.x * blockDim.x + threadIdx.x;
  if (idx >= E * H) return;
  const int e = idx / H, h = idx % H;
  if (!mask[e]) return;
  const float lg = leakyf(el[src[e] * H + h] + er[dst[e] * H + h], LRELU_ATT);
  atomicMax(&mint[dst[e] * H + h], f2o(lg));
}

// ---------------- self-loop p -> init s and acc -----------------------------
__global__ void gg_self_init(const float* __restrict__ Hf, const float* __restrict__ el,
                             const float* __restrict__ er, const int* __restrict__ mint,
                             float* __restrict__ s, float* __restrict__ acc,
                             int N, int H, int D) {
  const int idx = blockIdx.x * blockDim.x + threadIdx.x;
  if (idx >= N * H * D) return;
  const int n = idx / (H * D);
  const int hd = idx % (H * D);
  const int h = hd / D;
  const float m = o2f(mint[n * H + h]);
  const float p = __expf(leakyf(el[n * H + h] + er[n * H + h], LRELU_ATT) - m);
  acc[idx] = p * Hf[idx];
  if ((hd % D) == 0) s[n * H + h] = p;
}

// ---------------- segment sum of p over dst ---------------------------------
__global__ void gg_sum_edges(const float* __restrict__ el, const float* __restrict__ er,
                             const int* __restrict__ src, const int* __restrict__ dst,
                             const int* __restrict__ mask, const int* __restrict__ mint,
                             float* __restrict__ s, int E, int H) {
  const int idx = blockIdx.x * blockDim.x + threadIdx.x;
  if (idx >= E * H) return;
  const int e = idx / H, h = idx % H;
  if (!mask[e]) return;
  const int dn = dst[e];
  const float lg = leakyf(el[src[e] * H + h] + er[dn * H + h], LRELU_ATT);
  const float p = __expf(lg - o2f(mint[dn * H + h]));
  atomicAdd(&s[dn * H + h], p);
}

// ---------------- aggregation: acc[dst] += p * Hf[src]  (wave per edge) -----
__global__ void gg_agg_edges(const float* __restrict__ Hf, const float* __restrict__ el,
                             const float* __restrict__ er, const int* __restrict__ src,
                             const int* __restrict__ dst, const int* __restrict__ mask,
                             const int* __restrict__ mint, float* __restrict__ acc,
                             int E, int H, int D) {
  const int e = blockIdx.x * (blockDim.x >> 5) + (threadIdx.x >> 5);
  const int lane = threadIdx.x & 31;
  if (e >= E) return;
  if (!mask[e]) return;
  const int sn = src[e], dn = dst[e];
  const int HD = H * D;
  const int fpl = HD >> 5;                  // 8 (layer1) or 2 (layer2)
  const int f0 = lane * fpl;
  const int h = f0 / D;                     // all fpl features share one head here
  const float lg = leakyf(el[sn * H + h] + er[dn * H + h], LRELU_ATT);
  const float p = __expf(lg - o2f(mint[dn * H + h]));
  for (int j = 0; j < fpl; ++j)
    atomicAdd(&acc[(size_t)dn * HD + f0 + j], p * Hf[(size_t)sn * HD + f0 + j]);
}

// ---------------- finalize: out = acc / s + bias (optional leaky) -----------
__global__ void gg_finalize(const float* __restrict__ acc, const float* __restrict__ s,
                            const float* __restrict__ bias, float* __restrict__ out,
                            int N, int H, int D, int activate) {
  const int idx = blockIdx.x * blockDim.x + threadIdx.x;
  if (idx >= N * H * D) return;
  const int n = idx / (H * D);
  const int hd = idx % (H * D);
  const int h = hd / D;
  float v = acc[idx] / fmaxf(s[n * H + h], 1e-12f) + bias[hd];
  if (activate) v = leakyf(v, LRELU_ACT);
  out[idx] = v;
}

static inline int cdiv(int a, int b) { return (a + b - 1) / b; }

extern "C" void kernel_launch(void* const* d_in, const int* in_sizes, int n_in,
                              void* d_out, int out_size, void* d_ws, size_t ws_size,
                              hipStream_t stream) {
  const float* x   = (const float*)d_in[0];
  const int*   src = (const int*)d_in[1];
  const int*   dst = (const int*)d_in[2];
  const float* W1  = (const float*)d_in[3];
  const float* al1 = (const float*)d_in[4];
  const float* ar1 = (const float*)d_in[5];
  const float* b1  = (const float*)d_in[6];
  const float* W2  = (const float*)d_in[7];
  const float* al2 = (const float*)d_in[8];
  const float* ar2 = (const float*)d_in[9];
  const float* b2  = (const float*)d_in[10];
  float* out = (float*)d_out;

  const int FIN = 256, H1 = 4, D1 = 64, H2 = 16, D2 = 4;
  const int HD1 = H1 * D1;                  // 256
  const int HD2 = H2 * D2;                  // 64
  const int N = in_sizes[0] / FIN;          // 50000
  const int E = in_sizes[1];                // 800000

  // workspace layout (floats)
  float* hfeat = (float*)d_ws;                          // N*256  (hfeat1, then hfeat2)
  float* accb  = hfeat + (size_t)N * HD1;               // N*256  (acc1, then activated h1)
  float* el    = accb + (size_t)N * HD1;                // N*16
  float* er    = el + (size_t)N * H2;                   // N*16
  int*   mint  = (int*)(er + (size_t)N * H2);           // N*16
  float* sbuf  = (float*)(mint + (size_t)N * H2);       // N*16
  int*   mask  = (int*)(sbuf + (size_t)N * H2);         // E
  float* nrm   = (float*)(mask + (size_t)E);            // N

  const int TB = 256;
  const dim3 bGemm(128);

  // ===== layer 1 =====
  gg_row_norm<<<cdiv(N, 8), TB, 0, stream>>>(x, nrm, N, FIN);
  gg_edge_mask<<<cdiv(E, 8), TB, 0, stream>>>(x, nrm, src, dst, mask, E, FIN, 0);

  gg_gemm_f32_wmma<<<dim3(cdiv(N, 16), HD1 / 64), bGemm, 0, stream>>>(x, W1, hfeat, N, FIN, HD1);

  gg_logits<<<cdiv(N * H1, TB), TB, 0, stream>>>(hfeat, al1, ar1, el, er, mint, N, H1, D1);
  gg_max_edges<<<cdiv(E * H1, TB), TB, 0, stream>>>(el, er, src, dst, mask, mint, E, H1);
  gg_self_init<<<cdiv(N * HD1, TB), TB, 0, stream>>>(hfeat, el, er, mint, sbuf, accb, N, H1, D1);
  gg_sum_edges<<<cdiv(E * H1, TB), TB, 0, stream>>>(el, er, src, dst, mask, mint, sbuf, E, H1);
  gg_agg_edges<<<cdiv(E, 8), TB, 0, stream>>>(hfeat, el, er, src, dst, mask, mint, accb, E, H1, D1);
  gg_finalize<<<cdiv(N * HD1, TB), TB, 0, stream>>>(accb, sbuf, b1, accb, N, H1, D1, 1);

  // ===== layer 2 =====
  gg_row_norm<<<cdiv(N, 8), TB, 0, stream>>>(accb, nrm, N, HD1);
  gg_edge_mask<<<cdiv(E, 8), TB, 0, stream>>>(accb, nrm, src, dst, mask, E, HD1, 1);

  gg_gemm_f32_wmma<<<dim3(cdiv(N, 16), HD2 / 64), bGemm, 0, stream>>>(accb, W2, hfeat, N, HD1, HD2);

  gg_logits<<<cdiv(N * H2, TB), TB, 0, stream>>>(hfeat, al2, ar2, el, er, mint, N, H2, D2);
  gg_max_edges<<<cdiv(E * H2, TB), TB, 0, stream>>>(el, er, src, dst, mask, mint, E, H2);
  gg_self_init<<<cdiv(N * HD2, TB), TB, 0, stream>>>(hfeat, el, er, mint, sbuf, out, N, H2, D2);
  gg_sum_edges<<<cdiv(E * H2, TB), TB, 0, stream>>>(el, er, src, dst, mask, mint, sbuf, E, H2);
  gg_agg_edges<<<cdiv(E, 8), TB, 0, stream>>>(hfeat, el, er, src, dst, mask, mint, out, E, H2, D2);
  gg_finalize<<<cdiv(N * HD2, TB), TB, 0, stream>>>(out, sbuf, b2, out, N, H2, D2, 0);
}